// PointsetFeaturePropagation_6055903887567
// MI455X (gfx1250) — compile-verified
//
#include <hip/hip_runtime.h>

typedef __attribute__((ext_vector_type(2))) float v2f;
typedef __attribute__((ext_vector_type(8))) float v8f;

// Problem dims (from reference setup_inputs)
constexpr int B    = 4;
constexpr int N    = 16384;
constexpr int S    = 2048;
constexpr int D1   = 64;
constexpr int D2   = 256;
constexpr int CMID = 256;
constexpr int NCOL = B * N;       // 65536 columns
constexpr int KCAT = D1 + D2;     // 320

// ---------------------------------------------------------------------------
// Kernel 1: transpose feats2 [B,D2,S] -> f2t [B,S,D2] (gather-friendly rows)
// ---------------------------------------------------------------------------
__global__ __launch_bounds__(256) void k_transpose_f2(const float* __restrict__ feats2,
                                                      float* __restrict__ f2t) {
    int t = blockIdx.x * 256 + threadIdx.x;            // t = ((b*S)+s)*D2 + d
    int d = t & (D2 - 1);
    int rest = t >> 8;                                 // (b*S + s)
    int s = rest & (S - 1);
    int b = rest >> 11;
    f2t[t] = feats2[(b * D2 + d) * S + s];
}

// ---------------------------------------------------------------------------
// Kernel 2: brute-force 3-NN + inverse-distance weights
//   pos1 [B,3,N], pos2 [B,3,S] -> idx3 [B*N,3], w3 [B*N,3]
// ---------------------------------------------------------------------------
__global__ __launch_bounds__(256) void k_three_nn(const float* __restrict__ pos1,
                                                  const float* __restrict__ pos2,
                                                  int* __restrict__ idx3,
                                                  float* __restrict__ w3) {
    __shared__ float px[S], py[S], pz[S];
    const int b = blockIdx.y;
    const int tid = threadIdx.x;
    for (int i = tid; i < S; i += 256) {
        px[i] = pos2[(b * 3 + 0) * S + i];
        py[i] = pos2[(b * 3 + 1) * S + i];
        pz[i] = pos2[(b * 3 + 2) * S + i];
    }
    __syncthreads();

    const int n = blockIdx.x * 256 + tid;
    const float qx = pos1[(b * 3 + 0) * N + n];
    const float qy = pos1[(b * 3 + 1) * N + n];
    const float qz = pos1[(b * 3 + 2) * N + n];

    float d0 = 3.0e38f, d1 = 3.0e38f, d2 = 3.0e38f;
    int i0 = 0, i1 = 0, i2 = 0;
    for (int s = 0; s < S; ++s) {
        float dx = qx - px[s], dy = qy - py[s], dz = qz - pz[s];
        float d = dx * dx + dy * dy + dz * dz;
        if (d < d2) {
            if (d < d1) {
                d2 = d1; i2 = i1;
                if (d < d0) { d1 = d0; i1 = i0; d0 = d; i0 = s; }
                else        { d1 = d;  i1 = s; }
            } else { d2 = d; i2 = s; }
        }
    }
    float r0 = 1.0f / (d0 + 1e-8f);
    float r1 = 1.0f / (d1 + 1e-8f);
    float r2 = 1.0f / (d2 + 1e-8f);
    float inv = 1.0f / (r0 + r1 + r2);
    const int col = b * N + n;
    idx3[col * 3 + 0] = i0; idx3[col * 3 + 1] = i1; idx3[col * 3 + 2] = i2;
    w3[col * 3 + 0] = r0 * inv; w3[col * 3 + 1] = r1 * inv; w3[col * 3 + 2] = r2 * inv;
}

// ---------------------------------------------------------------------------
// Kernel 3: build xcat [KCAT x NCOL] row-major = [feats1 ; interp]
// ---------------------------------------------------------------------------
__global__ __launch_bounds__(256) void k_interp_concat(const float* __restrict__ feats1,
                                                       const float* __restrict__ f2t,
                                                       const int* __restrict__ idx3,
                                                       const float* __restrict__ w3,
                                                       float* __restrict__ xcat) {
    const int lane = threadIdx.x & 31;
    const int wv = threadIdx.x >> 5;                   // 0..7
    const int col = blockIdx.x * 32 + lane;
    const int b = col >> 14;                           // col / N
    const int n = col & (N - 1);

    // rows 0..63: copy feats1
    #pragma unroll
    for (int j = 0; j < 8; ++j) {
        int c = wv * 8 + j;
        xcat[c * NCOL + col] = feats1[(b * D1 + c) * N + n];
    }

    // rows 64..319: 3-NN weighted interpolation of feats2
    const int i0 = idx3[col * 3 + 0];
    const int i1 = idx3[col * 3 + 1];
    const int i2 = idx3[col * 3 + 2];
    const float wq0 = w3[col * 3 + 0];
    const float wq1 = w3[col * 3 + 1];
    const float wq2 = w3[col * 3 + 2];
    const float* base = f2t + (size_t)b * S * D2;
    const float* r0 = base + i0 * D2;
    const float* r1 = base + i1 * D2;
    const float* r2 = base + i2 * D2;
    #pragma unroll 4
    for (int j = 0; j < 32; ++j) {
        int d = wv * 32 + j;
        float v = wq0 * r0[d] + wq1 * r1[d] + wq2 * r2[d];
        xcat[(D1 + d) * NCOL + col] = v;
    }
}

// ---------------------------------------------------------------------------
// Kernel 4: GEMM  Y[M=256 x NCOL] = W[256 x K] @ X[K x NCOL] + bias
//   One wave per 64x64 macro-tile = 4x4 grid of 16x16 WMMA accumulators.
//   Per K-step of 4: 4 A v2f loads + 8 B b32 loads feed 16 v_wmma -> 4B/lane/wmma.
//   Operand placement per ISA layout tables:
//     A: [lane&15][k0 + r + (lane>>4)*2]   B: [k0 + r + (lane>>4)*2][col + lane&15]
//     D: M = r + (lane>>4)*8, Ncol = lane&15
// ---------------------------------------------------------------------------
__global__ __launch_bounds__(256) void k_gemm_wmma(const float* __restrict__ W,
                                                   const float* __restrict__ X,
                                                   const float* __restrict__ bias,
                                                   float* __restrict__ Y,
                                                   int K) {
    const int lane = threadIdx.x & 31;
    const int wave = threadIdx.x >> 5;
    const int job  = blockIdx.x * 8 + wave;            // 0 .. 4095
    const int jobM = job & 3;                          // 4 M-strips of 64
    const int jobC = job >> 2;                         // 1024 col-strips of 64
    const int ln = lane & 15;
    const int half = lane >> 4;                        // 0 or 1
    const int kh = half * 2;
    const int m0 = jobM * 64;
    const int col0 = jobC * 64;

    const float* Arow[4];
    const float* Bcol[4];
    #pragma unroll
    for (int i = 0; i < 4; ++i) Arow[i] = W + (m0 + i * 16 + ln) * K + kh;
    #pragma unroll
    for (int j = 0; j < 4; ++j) Bcol[j] = X + col0 + j * 16 + ln;

    v8f acc[4][4] = {};
    for (int k0 = 0; k0 < K; k0 += 4) {
        v2f a[4];
        v2f bb[4];
        #pragma unroll
        for (int i = 0; i < 4; ++i) a[i] = *(const v2f*)(Arow[i] + k0);
        #pragma unroll
        for (int j = 0; j < 4; ++j) {
            bb[j][0] = Bcol[j][(k0 + kh + 0) * NCOL];
            bb[j][1] = Bcol[j][(k0 + kh + 1) * NCOL];
        }
        #pragma unroll
        for (int i = 0; i < 4; ++i)
            #pragma unroll
            for (int j = 0; j < 4; ++j)
                acc[i][j] = __builtin_amdgcn_wmma_f32_16x16x4_f32(
                    /*neg_a=*/false, a[i], /*neg_b=*/false, bb[j],
                    /*c_mod=*/(short)0, acc[i][j],
                    /*reuse_a=*/false, /*reuse_b=*/false);
    }

    #pragma unroll
    for (int i = 0; i < 4; ++i) {
        float bv[8];
        #pragma unroll
        for (int r = 0; r < 8; ++r) bv[r] = bias[m0 + i * 16 + half * 8 + r];
        #pragma unroll
        for (int j = 0; j < 4; ++j) {
            const int colOut = col0 + j * 16 + ln;
            #pragma unroll
            for (int r = 0; r < 8; ++r) {
                int m = m0 + i * 16 + half * 8 + r;
                Y[(size_t)m * NCOL + colOut] = acc[i][j][r] + bv[r];
            }
        }
    }
}

// ---------------------------------------------------------------------------
// Kernel 5: per-channel BN statistics (training mode: over B,N = all NCOL)
// ---------------------------------------------------------------------------
__global__ __launch_bounds__(256) void k_bn_stats(const float* __restrict__ Y,
                                                  const float* __restrict__ g,
                                                  const float* __restrict__ be,
                                                  float* __restrict__ coefA,
                                                  float* __restrict__ coefB) {
    __shared__ float ls[256], lq[256];
    const int o = blockIdx.x;
    const int tid = threadIdx.x;
    const float* row = Y + (size_t)o * NCOL;
    float s = 0.0f, q = 0.0f;
    for (int i = tid; i < NCOL; i += 256) {
        float v = row[i];
        s += v; q += v * v;
    }
    ls[tid] = s; lq[tid] = q;
    __syncthreads();
    for (int off = 128; off > 0; off >>= 1) {
        if (tid < off) { ls[tid] += ls[tid + off]; lq[tid] += lq[tid + off]; }
        __syncthreads();
    }
    if (tid == 0) {
        float mean = ls[0] * (1.0f / NCOL);
        float var = lq[0] * (1.0f / NCOL) - mean * mean;
        float rs = rsqrtf(var + 1e-5f);
        float a = rs * g[o];
        coefA[o] = a;
        coefB[o] = be[o] - mean * a;
    }
}

// ---------------------------------------------------------------------------
// Kernel 6: BN apply + ReLU. remap==0: in-place on [M x NCOL].
//   remap==1: write to out[b][o][n] (reference output layout).
// ---------------------------------------------------------------------------
__global__ __launch_bounds__(256) void k_bn_relu(float* __restrict__ Y,
                                                 const float* __restrict__ coefA,
                                                 const float* __restrict__ coefB,
                                                 float* __restrict__ out,
                                                 int remap) {
    const int t = blockIdx.x * 256 + threadIdx.x;      // t = o*NCOL + col
    const int o = t >> 16;
    const int col = t & (NCOL - 1);
    float v = fmaf(Y[t], coefA[o], coefB[o]);
    v = fmaxf(v, 0.0f);
    if (remap) {
        int b = col >> 14;
        int n = col & (N - 1);
        out[((size_t)(b * CMID + o)) * N + n] = v;
    } else {
        Y[t] = v;
    }
}

// ---------------------------------------------------------------------------
extern "C" void kernel_launch(void* const* d_in, const int* in_sizes, int n_in,
                              void* d_out, int out_size, void* d_ws, size_t ws_size,
                              hipStream_t stream) {
    const float* pos1   = (const float*)d_in[0];
    const float* pos2   = (const float*)d_in[1];
    const float* feats1 = (const float*)d_in[2];
    const float* feats2 = (const float*)d_in[3];
    const float* w0 = (const float*)d_in[4];
    const float* b0 = (const float*)d_in[5];
    const float* g0 = (const float*)d_in[6];
    const float* be0 = (const float*)d_in[7];
    const float* w1 = (const float*)d_in[8];
    const float* b1 = (const float*)d_in[9];
    const float* g1 = (const float*)d_in[10];
    const float* be1 = (const float*)d_in[11];

    char* ws = (char*)d_ws;
    size_t off = 0;
    int*   idx3 = (int*)(ws + off);   off += (size_t)NCOL * 3 * sizeof(int);     // 0.79 MB
    float* w3   = (float*)(ws + off); off += (size_t)NCOL * 3 * sizeof(float);   // 0.79 MB
    float* f2t  = (float*)(ws + off); off += (size_t)B * S * D2 * sizeof(float); // 8 MB
    float* y0   = (float*)(ws + off); off += (size_t)CMID * NCOL * sizeof(float);// 67 MB
    float* xcat = (float*)(ws + off); off += (size_t)KCAT * NCOL * sizeof(float);// 84 MB
    float* y1   = xcat;  // GEMM2 no longer needs xcat; reuse (84 MB >= 67 MB)
    float* coefA0 = (float*)(ws + off); off += CMID * sizeof(float);
    float* coefB0 = (float*)(ws + off); off += CMID * sizeof(float);
    float* coefA1 = (float*)(ws + off); off += CMID * sizeof(float);
    float* coefB1 = (float*)(ws + off); off += CMID * sizeof(float);

    // 1) transpose feats2 -> f2t
    k_transpose_f2<<<(B * S * D2) / 256, 256, 0, stream>>>(feats2, f2t);

    // 2) 3-NN + weights
    dim3 g2(N / 256, B);
    k_three_nn<<<g2, 256, 0, stream>>>(pos1, pos2, idx3, w3);

    // 3) interpolate + concat -> xcat [320 x 65536]
    k_interp_concat<<<NCOL / 32, 256, 0, stream>>>(feats1, f2t, idx3, w3, xcat);

    // 4) layer 0: y0 = w0 @ xcat + b0   (WMMA f32, 64x64 per wave)
    k_gemm_wmma<<<((CMID / 64) * (NCOL / 64)) / 8, 256, 0, stream>>>(w0, xcat, b0, y0, KCAT);

    // 5) BN stats + apply + ReLU (in place)
    k_bn_stats<<<CMID, 256, 0, stream>>>(y0, g0, be0, coefA0, coefB0);
    k_bn_relu<<<(CMID * NCOL) / 256, 256, 0, stream>>>(y0, coefA0, coefB0, y0, 0);

    // 6) layer 1: y1 = w1 @ y0 + b1   (WMMA f32, 64x64 per wave)
    k_gemm_wmma<<<((CMID / 64) * (NCOL / 64)) / 8, 256, 0, stream>>>(w1, y0, b1, y1, CMID);

    // 7) BN stats + apply + ReLU -> d_out with [b,o,n] layout
    k_bn_stats<<<CMID, 256, 0, stream>>>(y1, g1, be1, coefA1, coefB1);
    k_bn_relu<<<(CMID * NCOL) / 256, 256, 0, stream>>>(y1, coefA1, coefB1, (float*)d_out, 1);
}